// RGCN_87471303950603
// MI455X (gfx1250) — compile-verified
//
#include <hip/hip_runtime.h>

#define NPAT 50000
#define NCON 20000
#define NNODE 70000     // NPAT + NCON, multiple of 16
#define EDGES 800000
#define HD 128
#define DP 64
#define DC 128
#define NSTRIP (NNODE / 16)

typedef float v2f __attribute__((ext_vector_type(2)));
typedef float v8f __attribute__((ext_vector_type(8)));

// Stage W[K,HD] (row-major, global) into LDS pre-swizzled for WMMA B fragments:
//   Bsw[P*256 + 2*c + 0] = W[2P  ][c]
//   Bsw[P*256 + 2*c + 1] = W[2P+1][c]     (P = K-pair index, c = column)
// so a lane's B fragment (B[kA][c], B[kA+1][c]) is one aligned 8-byte LDS load.
// Two float4 row reads -> two contiguous ds_store_b128 per task.
__device__ __forceinline__ void stage_B_swz(const float* __restrict__ W, float* Bs,
                                            int K, int tid) {
  const int ntask = (K / 2) * (HD / 4);  // pairs x col-quads
  for (int t = tid; t < ntask; t += 256) {
    const int P = t >> 5;
    const int c4 = (t & 31) * 4;
    const float4 r0 = *(const float4*)(W + (size_t)(2 * P) * HD + c4);
    const float4 r1 = *(const float4*)(W + (size_t)(2 * P + 1) * HD + c4);
    float4* dst = (float4*)(Bs + P * 256 + c4 * 2);
    dst[0] = float4{r0.x, r1.x, r0.y, r1.y};
    dst[1] = float4{r0.z, r1.z, r0.w, r1.w};
  }
}

// Accumulate a 16-row strip (rows arow..arow+15 of A, all 8 column tiles of B)
// into acc[8] using V_WMMA_F32_16X16X4_F32. A: [*, K] row-major from global
// (float2 per lane); B fragments: single b64 LDS loads from the swizzled stage.
// Per ISA 7.12.2: A f32 16x4 -> lanes 0-15 hold K={kk,kk+1}, lanes 16-31 K={kk+2,kk+3};
// B 4x16 -> one row per VGPR striped across lanes, K split across lane halves.
__device__ __forceinline__ void wmma_strip(const float* __restrict__ A, int arow, int K,
                                           const float* Bs, v8f* acc,
                                           int mrow, int half, int ncol) {
  for (int kk = 0; kk < K; kk += 4) {
    const v2f a = *(const v2f*)(A + (size_t)(arow + mrow) * K + kk + 2 * half);
    const float* b0 = Bs + (kk / 2 + half) * 256 + 2 * ncol;
#pragma unroll
    for (int nt = 0; nt < 8; ++nt) {
      const v2f b = *(const v2f*)(b0 + nt * 32);  // contiguous, 8B-aligned
      acc[nt] = __builtin_amdgcn_wmma_f32_16x16x4_f32(
          false, a, false, b, (short)0, acc[nt], false, false);
    }
  }
}

// Per-type projection GEMM: out[out_row0 + i] = A[i] @ W + bias, W staged in LDS.
__global__ void __launch_bounds__(256)
proj_kernel(const float* __restrict__ A, const float* __restrict__ W,
            const float* __restrict__ bias, int K, int nstrips, int out_row0,
            float* __restrict__ out) {
  __shared__ float Bs[DC * HD];  // 64 KB, covers K in {64,128}
  const int tid = threadIdx.x;
  stage_B_swz(W, Bs, K, tid);
  __syncthreads();

  const int strip = blockIdx.x * 8 + (tid >> 5);  // wave-uniform
  if (strip >= nstrips) return;                   // whole-wave exit, no later barriers
  const int lane = tid & 31;
  const int mrow = lane & 15, half = lane >> 4, ncol = lane & 15;

  v8f acc[8];
  const v8f z = {0.f, 0.f, 0.f, 0.f, 0.f, 0.f, 0.f, 0.f};
#pragma unroll
  for (int nt = 0; nt < 8; ++nt) acc[nt] = z;

  wmma_strip(A, strip * 16, K, Bs, acc, mrow, half, ncol);

  const int orow = out_row0 + strip * 16;
#pragma unroll
  for (int nt = 0; nt < 8; ++nt) {
    const float bv = bias[nt * 16 + ncol];
#pragma unroll
    for (int i = 0; i < 8; ++i)
      out[(size_t)(orow + i + 8 * half) * HD + nt * 16 + ncol] = acc[nt][i] + bv;
  }
}

// Fused layer: out = relu(x@Wroot + broot + agg0@Wrel0 + agg1@Wrel1),
// each 64 KB weight matrix staged (swizzled) into LDS in turn.
__global__ void __launch_bounds__(256)
rgcn_layer_kernel(const float* __restrict__ x, const float* __restrict__ agg0,
                  const float* __restrict__ agg1, const float* __restrict__ Wroot,
                  const float* __restrict__ broot, const float* __restrict__ Wrel0,
                  const float* __restrict__ Wrel1, float* __restrict__ out) {
  __shared__ float Bs[HD * HD];  // 64 KB
  const int tid = threadIdx.x;
  const int lane = tid & 31;
  const int mrow = lane & 15, half = lane >> 4, ncol = lane & 15;
  const int strip = blockIdx.x * 8 + (tid >> 5);  // wave-uniform
  const bool active = (strip < NSTRIP);
  const int m0 = strip * 16;

  v8f acc[8];
  const v8f z = {0.f, 0.f, 0.f, 0.f, 0.f, 0.f, 0.f, 0.f};
#pragma unroll
  for (int nt = 0; nt < 8; ++nt) acc[nt] = z;

  const float* Am[3] = {x, agg0, agg1};
  const float* Bm[3] = {Wroot, Wrel0, Wrel1};
#pragma unroll
  for (int p = 0; p < 3; ++p) {
    __syncthreads();  // previous pair's LDS reads done before overwrite
    stage_B_swz(Bm[p], Bs, HD, tid);
    __syncthreads();
    if (active)  // wave-uniform: EXEC all-ones at WMMA
      wmma_strip(Am[p], m0, HD, Bs, acc, mrow, half, ncol);
  }

  if (active) {
#pragma unroll
    for (int nt = 0; nt < 8; ++nt) {
      const float bv = broot[nt * 16 + ncol];
#pragma unroll
      for (int i = 0; i < 8; ++i) {
        const float v = acc[nt][i] + bv;
        out[(size_t)(m0 + i + 8 * half) * HD + nt * 16 + ncol] = fmaxf(v, 0.f);
      }
    }
  }
}

// ---- CSR build (once per call; edge lists are launch-invariant) ----

__global__ void __launch_bounds__(256)
count_kernel(const int* __restrict__ dst_pc, const int* __restrict__ dst_cp,
             int* __restrict__ cnt0, int* __restrict__ cnt1) {
  const int e = blockIdx.x * blockDim.x + threadIdx.x;
  if (e >= EDGES) return;
  atomicAdd(&cnt0[NPAT + dst_pc[e]], 1);
  atomicAdd(&cnt1[dst_cp[e]], 1);
}

// One block per relation: exclusive scan of counts -> offsets + fill cursors,
// plus reciprocal mean normalizers 1/max(deg,1).
__global__ void __launch_bounds__(1024)
scan_kernel(const int* __restrict__ cnt0, int* __restrict__ off0, int* __restrict__ cur0,
            float* __restrict__ inv0, const int* __restrict__ cnt1,
            int* __restrict__ off1, int* __restrict__ cur1, float* __restrict__ inv1) {
  const int* cnt = (blockIdx.x == 0) ? cnt0 : cnt1;
  int* off = (blockIdx.x == 0) ? off0 : off1;
  int* cur = (blockIdx.x == 0) ? cur0 : cur1;
  float* inv = (blockIdx.x == 0) ? inv0 : inv1;

  __shared__ int sdata[1024];
  __shared__ int srun;
  const int tid = threadIdx.x;
  if (tid == 0) srun = 0;
  __syncthreads();

  for (int base = 0; base < NNODE; base += 1024) {
    const int i = base + tid;
    const int v = (i < NNODE) ? cnt[i] : 0;
    sdata[tid] = v;
    __syncthreads();
    for (int d = 1; d < 1024; d <<= 1) {
      const int t = (tid >= d) ? sdata[tid - d] : 0;
      __syncthreads();
      sdata[tid] += t;
      __syncthreads();
    }
    const int incl = sdata[tid];
    const int run = srun;  // uniform
    if (i < NNODE) {
      const int excl = run + incl - v;
      off[i] = excl;
      cur[i] = excl;
      inv[i] = 1.0f / fmaxf((float)v, 1.0f);
    }
    __syncthreads();
    if (tid == 1023) srun = run + incl;  // incl of last thread == chunk total
    __syncthreads();
  }
  if (tid == 0) off[NNODE] = srun;
}

__global__ void __launch_bounds__(256)
fill_kernel(const int* __restrict__ src_pc, const int* __restrict__ dst_pc,
            const int* __restrict__ src_cp, const int* __restrict__ dst_cp,
            int* __restrict__ cur0, int* __restrict__ cur1,
            int* __restrict__ csr0, int* __restrict__ csr1) {
  const int e = blockIdx.x * blockDim.x + threadIdx.x;
  if (e >= EDGES) return;
  const int p = atomicAdd(&cur0[NPAT + dst_pc[e]], 1);
  csr0[p] = src_pc[e];                 // source row = patient id
  const int q = atomicAdd(&cur1[dst_cp[e]], 1);
  csr1[q] = NPAT + src_cp[e];          // source row = concept id + NPAT
}

// Atomic-free mean aggregation: one wave per destination row, lane = float4 chunk.
// agg[d] = inv[d] * sum over incoming edges of x[src]. Zero-degree rows -> 0.
__global__ void __launch_bounds__(256)
gather_kernel(const float* __restrict__ x, const int* __restrict__ csr,
              const int* __restrict__ off, const float* __restrict__ inv,
              float* __restrict__ agg) {
  const int d = (blockIdx.x * blockDim.x + threadIdx.x) >> 5;  // wave-uniform
  if (d >= NNODE) return;
  const int h = (threadIdx.x & 31) * 4;
  const int beg = off[d], end = off[d + 1];

  float4 acc = {0.f, 0.f, 0.f, 0.f};
  int j = beg;
  for (; j + 3 < end; j += 4) {  // 4-way unroll: 4 independent 16B gathers in flight
    const int s0 = csr[j], s1 = csr[j + 1], s2 = csr[j + 2], s3 = csr[j + 3];
    const float4 v0 = *(const float4*)(x + (size_t)s0 * HD + h);
    const float4 v1 = *(const float4*)(x + (size_t)s1 * HD + h);
    const float4 v2 = *(const float4*)(x + (size_t)s2 * HD + h);
    const float4 v3 = *(const float4*)(x + (size_t)s3 * HD + h);
    acc.x += v0.x + v1.x + v2.x + v3.x;
    acc.y += v0.y + v1.y + v2.y + v3.y;
    acc.z += v0.z + v1.z + v2.z + v3.z;
    acc.w += v0.w + v1.w + v2.w + v3.w;
  }
  for (; j < end; ++j) {
    const int s = csr[j];
    const float4 v = *(const float4*)(x + (size_t)s * HD + h);
    acc.x += v.x; acc.y += v.y; acc.z += v.z; acc.w += v.w;
  }
  const float sc = inv[d];
  const float4 o = {acc.x * sc, acc.y * sc, acc.z * sc, acc.w * sc};
  *(float4*)(agg + (size_t)d * HD + h) = o;
}

extern "C" void kernel_launch(void* const* d_in, const int* in_sizes, int n_in,
                              void* d_out, int out_size, void* d_ws, size_t ws_size,
                              hipStream_t stream) {
  const float* xp    = (const float*)d_in[0];
  const float* xc    = (const float*)d_in[1];
  const float* Wp    = (const float*)d_in[2];
  const float* bp    = (const float*)d_in[3];
  const float* Wc    = (const float*)d_in[4];
  const float* bc    = (const float*)d_in[5];
  const float* Wroot = (const float*)d_in[6];
  const float* broot = (const float*)d_in[7];
  const float* Wrel  = (const float*)d_in[8];
  const int* src_pc  = (const int*)d_in[9];
  const int* dst_pc  = (const int*)d_in[10];
  const int* src_cp  = (const int*)d_in[11];
  const int* dst_cp  = (const int*)d_in[12];

  // Workspace layout (16B-aligned sections, float4-accessed arrays first).
  const size_t NH = (size_t)NNODE * HD;
  float* xA   = (float*)d_ws;
  float* xB   = xA + NH;
  float* agg0 = xB + NH;
  float* agg1 = agg0 + NH;
  float* inv0 = agg1 + NH;
  float* inv1 = inv0 + NNODE;
  int*   cnt0 = (int*)(inv1 + NNODE);
  int*   cnt1 = cnt0 + NNODE;
  int*   off0 = cnt1 + NNODE;
  int*   off1 = off0 + (NNODE + 8);
  int*   cur0 = off1 + (NNODE + 8);
  int*   cur1 = cur0 + NNODE;
  int*   csr0 = cur1 + NNODE;
  int*   csr1 = csr0 + EDGES;

  const int eblk = (EDGES + 255) / 256;

  // CSR build (layer-invariant). csr bucket order is atomic-race dependent, which
  // only permutes fp addition order inside each destination's mean (tolerance-level).
  hipMemsetAsync(cnt0, 0, 2 * (size_t)NNODE * sizeof(int), stream);
  count_kernel<<<eblk, 256, 0, stream>>>(dst_pc, dst_cp, cnt0, cnt1);
  scan_kernel<<<2, 1024, 0, stream>>>(cnt0, off0, cur0, inv0, cnt1, off1, cur1, inv1);
  fill_kernel<<<eblk, 256, 0, stream>>>(src_pc, dst_pc, src_cp, dst_cp,
                                        cur0, cur1, csr0, csr1);

  // Initial per-type projections (WMMA, weights in LDS).
  proj_kernel<<<(3125 + 7) / 8, 256, 0, stream>>>(xp, Wp, bp, DP, NPAT / 16, 0, xA);
  proj_kernel<<<(1250 + 7) / 8, 256, 0, stream>>>(xc, Wc, bc, DC, NCON / 16, NPAT, xA);

  const int gthr_blocks = NNODE / 8;                 // one wave per node row
  const int gemm_blocks = (NSTRIP + 7) / 8;          // 8 strips per 256-thread block

  const float* xin = xA;
  for (int l = 0; l < 2; ++l) {
    gather_kernel<<<gthr_blocks, 256, 0, stream>>>(xin, csr0, off0, inv0, agg0);
    gather_kernel<<<gthr_blocks, 256, 0, stream>>>(xin, csr1, off1, inv1, agg1);

    float* xout = (l == 0) ? xB : (float*)d_out;
    const float* Wr = Wroot + (size_t)l * HD * HD;
    const float* br = broot + (size_t)l * HD;
    const float* W0 = Wrel + (size_t)(2 * l + 0) * HD * HD;
    const float* W1 = Wrel + (size_t)(2 * l + 1) * HD * HD;
    rgcn_layer_kernel<<<gemm_blocks, 256, 0, stream>>>(xin, agg0, agg1, Wr, br, W0, W1, xout);
    xin = xout;
  }
}